// SSIMLoss_87943750353190
// MI455X (gfx1250) — compile-verified
//
#include <hip/hip_runtime.h>

typedef __attribute__((ext_vector_type(2))) float v2f;
typedef __attribute__((ext_vector_type(8))) float v8f;

#define IMG_H 512
#define IMG_W 512
#define PLANES 96          // 32 batch * 3 channels
#define TILE_W 64
#define TILE_H 32
#define HALO 5
#define IN_ROWS (TILE_H + 2*HALO)   // 42
#define IN_COLS (TILE_W + 2*HALO)   // 74
#define LDS_STRIDE 80               // padded: B-operand rows r / r+2 land in disjoint banks
#define NTHREADS 256
#define BLOCKS_X (IMG_W / TILE_W)   // 8
#define BLOCKS_Y (IMG_H / TILE_H)   // 16
#define NBLOCKS (PLANES * BLOCKS_X * BLOCKS_Y)  // 12288
#define C1F 0.0001f
#define C2F 0.0009f

// 11-tap Gaussian, sigma=1.5, normalized (matches reference _create_window row/col vector)
#define G0  0.00102838f
#define G1  0.00759876f
#define G2  0.03600077f
#define G3  0.10936069f
#define G4  0.21300553f
#define G5  0.26601172f

__device__ __forceinline__ float gauss_tap(int d) {
    // banded weight: g[d] for d in [0,10], else 0. Select chain (runs once per lane).
    const float G[11] = {G0,G1,G2,G3,G4,G5,G4,G3,G2,G1,G0};
    float v = 0.0f;
#pragma unroll
    for (int j = 0; j < 11; ++j) v = (d == j) ? G[j] : v;
    return v;
}

__global__ __launch_bounds__(NTHREADS) void ssim_tile_kernel(
    const float* __restrict__ pred, const float* __restrict__ target,
    float* __restrict__ partial) {
    __shared__ float sP[IN_ROWS * LDS_STRIDE];
    __shared__ float sT[IN_ROWS * LDS_STRIDE];
    __shared__ float sH[5][IN_ROWS * LDS_STRIDE];
    __shared__ float sWave[NTHREADS / 32];

    const int tid   = threadIdx.x;
    const int bid   = blockIdx.x;
    const int plane = bid >> 7;            // /128 tiles per plane
    const int rem   = bid & 127;
    const int X0    = (rem & (BLOCKS_X - 1)) * TILE_W;
    const int Y0    = (rem >> 3) * TILE_H;

    const float* p0 = pred   + (size_t)plane * (IMG_H * IMG_W);
    const float* t0 = target + (size_t)plane * (IMG_H * IMG_W);

    // ---- Stage inputs with zero padding (conv uses zero-pad 'same') ----
    for (int i = tid; i < IN_ROWS * IN_COLS; i += NTHREADS) {
        int r  = i / IN_COLS;
        int c  = i - r * IN_COLS;
        int gy = Y0 - HALO + r;
        int gx = X0 - HALO + c;
        float pv = 0.0f, tv = 0.0f;
        if ((unsigned)gy < IMG_H && (unsigned)gx < IMG_W) {
            size_t off = (size_t)gy * IMG_W + gx;
            pv = p0[off];
            tv = t0[off];
        }
        sP[r * LDS_STRIDE + c] = pv;
        sT[r * LDS_STRIDE + c] = tv;
    }
    __syncthreads();

    // ---- Horizontal 11-tap pass: 5 fields (p, t, p*p, t*t, p*t) ----
    {
        const float G[11] = {G0,G1,G2,G3,G4,G5,G4,G3,G2,G1,G0};
        for (int i = tid; i < IN_ROWS * TILE_W; i += NTHREADS) {
            int r = i >> 6;          // TILE_W == 64
            int x = i & 63;
            int base = r * LDS_STRIDE + x;
            float hp = 0.f, ht = 0.f, hpp = 0.f, htt = 0.f, hpt = 0.f;
#pragma unroll
            for (int k = 0; k < 11; ++k) {
                float pv = sP[base + k];
                float tv = sT[base + k];
                float g  = G[k];
                float gp = g * pv;
                float gt = g * tv;
                hp  = fmaf(g,  pv, hp);
                ht  = fmaf(g,  tv, ht);
                hpp = fmaf(gp, pv, hpp);
                htt = fmaf(gt, tv, htt);
                hpt = fmaf(gp, tv, hpt);
            }
            sH[0][base] = hp;
            sH[1][base] = ht;
            sH[2][base] = hpp;
            sH[3][base] = htt;
            sH[4][base] = hpt;
        }
    }
    __syncthreads();

    // ---- Vertical pass via WMMA: D[16x16] = W_band[16x26] x H[26x16] ----
    // A layout (16x4 f32): lane = M (0..15 mirrored), VGPR0/1 = K pair; lanes16-31 hold K=2,3.
    // B layout (4x16 f32): VGPR0 = rows K0(lanes0-15)/K2(lanes16-31), VGPR1 = K1/K3, N = lane&15.
    const int lane  = tid & 31;
    const int wv    = tid >> 5;            // 0..7 : one 16x16 tile per wave
    const int m     = lane & 15;
    const int khalf = lane >> 4;
    const int x0w   = (wv & 3) * 16;
    const int y0w   = (wv >> 2) * 16;

    v2f aW[7];
#pragma unroll
    for (int c = 0; c < 7; ++c) {
        int r0 = 4 * c + 2 * khalf;
        aW[c].x = gauss_tap(r0 - m);       // zero outside the band
        aW[c].y = gauss_tap(r0 + 1 - m);
    }

    v8f acc[5];
#pragma unroll
    for (int f = 0; f < 5; ++f) {
        v8f cacc = {0.f,0.f,0.f,0.f,0.f,0.f,0.f,0.f};
#pragma unroll
        for (int c = 0; c < 7; ++c) {
            int r0   = 4 * c + 2 * khalf;
            int row0 = y0w + r0;
            int row1 = y0w + r0 + 1;
            if (row0 > IN_ROWS - 1) row0 = IN_ROWS - 1;  // weights are 0 there
            if (row1 > IN_ROWS - 1) row1 = IN_ROWS - 1;
            v2f b;
            b.x = sH[f][row0 * LDS_STRIDE + x0w + (lane & 15)];
            b.y = sH[f][row1 * LDS_STRIDE + x0w + (lane & 15)];
            cacc = __builtin_amdgcn_wmma_f32_16x16x4_f32(
                false, aW[c], false, b, (short)0, cacc, false, false);
        }
        acc[f] = cacc;
    }

    // ---- SSIM map on the D-tile layout, then reduce ----
    float lsum = 0.0f;
#pragma unroll
    for (int v = 0; v < 8; ++v) {
        float mu1 = acc[0][v];
        float mu2 = acc[1][v];
        float ep  = acc[2][v];
        float et  = acc[3][v];
        float ept = acc[4][v];
        float mu1sq = mu1 * mu1;
        float mu2sq = mu2 * mu2;
        float mu12  = mu1 * mu2;
        float s1  = ep  - mu1sq;
        float s2  = et  - mu2sq;
        float s12 = ept - mu12;
        float num = (2.0f * mu12 + C1F) * (2.0f * s12 + C2F);
        float den = (mu1sq + mu2sq + C1F) * (s1 + s2 + C2F);
        lsum += num / den;
    }
#pragma unroll
    for (int off = 16; off > 0; off >>= 1)
        lsum += __shfl_xor(lsum, off, 32);
    if (lane == 0) sWave[wv] = lsum;
    __syncthreads();
    if (tid == 0) {
        float s = 0.0f;
#pragma unroll
        for (int i = 0; i < NTHREADS / 32; ++i) s += sWave[i];
        partial[bid] = s;
    }
}

__global__ __launch_bounds__(256) void ssim_reduce_kernel(
    const float* __restrict__ partial, float* __restrict__ out) {
    __shared__ float s[256];
    float acc = 0.0f;
    for (int i = threadIdx.x; i < NBLOCKS; i += 256) acc += partial[i];
    s[threadIdx.x] = acc;
    __syncthreads();
    for (int st = 128; st > 0; st >>= 1) {
        if (threadIdx.x < st) s[threadIdx.x] += s[threadIdx.x + st];
        __syncthreads();
    }
    if (threadIdx.x == 0) {
        const float invN = 1.0f / 25165824.0f;   // 32*3*512*512
        out[0] = 1.0f - s[0] * invN;
    }
}

extern "C" void kernel_launch(void* const* d_in, const int* in_sizes, int n_in,
                              void* d_out, int out_size, void* d_ws, size_t ws_size,
                              hipStream_t stream) {
    (void)in_sizes; (void)n_in; (void)out_size; (void)ws_size;
    const float* pred   = (const float*)d_in[0];
    const float* target = (const float*)d_in[1];
    float* out     = (float*)d_out;
    float* partial = (float*)d_ws;   // NBLOCKS floats of scratch

    ssim_tile_kernel<<<NBLOCKS, NTHREADS, 0, stream>>>(pred, target, partial);
    ssim_reduce_kernel<<<1, 256, 0, stream>>>(partial, out);
}